// PyTorchSourceModule_65936337928769
// MI455X (gfx1250) — compile-verified
//
#include <hip/hip_runtime.h>
#include <stdint.h>

// ---------------------------------------------------------------------------
// Harmonic sine source (NSF-style) for MI455X / gfx1250.
//   f0  : [32, 800]  frame-rate F0
//   W   : [9, 1]     projection weights
//   b   : [1]        bias
//   upp : 240        (hardcoded per reference; passed as device scalar)
// Outputs (concat, float32): har [32*192000], noise [32*192000], uv [32*192000]
//
// Store-bandwidth-bound design:
//  * 1 v_sin + 1 v_cos per sample; harmonics 2..9 via Chebyshev recurrence.
//  * v_sin/v_cos consume phase in revolutions -> no 2*pi multiply, no range
//    reduction (|phase| <= ~72 revolutions, well inside HW domain).
//  * tanh via v_exp_f32 + v_rcp_f32.
//  * Noise: PCG hash + Box-Muller (deterministic; matches the reference's
//    distribution, not its exact threefry bit-stream).
//  * CDNA5 path: Tensor Data Mover stages the 800-float f0 row into LDS
//    (tensor_load_to_lds + s_wait_tensorcnt), wave32 shuffle scan for the
//    phase prefix, float4 (b128) stores for all three outputs.
// ---------------------------------------------------------------------------

#define SR_INV        (1.0f / 24000.0f)
#define FRAME_UP      240
#define FRAMES        800
#define BATCH         32
#define SAMPLES       (FRAMES * FRAME_UP)      // 192000
#define STOT          (BATCH * SAMPLES)        // 6144000
#define SINE_AMP      0.1f
#define NOISE_STD     (0.1f / 3.0f)

#define FR_PER_BLK    32
#define SAMP_PER_BLK  (FR_PER_BLK * FRAME_UP)  // 7680
#define TPB           192                      // 6 wave32
#define ITERS         (SAMP_PER_BLK / (TPB * 4)) // 10, float4 per thread/iter

typedef uint32_t v4u __attribute__((ext_vector_type(4)));
typedef int      v4i __attribute__((ext_vector_type(4)));
typedef int      v8i __attribute__((ext_vector_type(8)));

#if defined(__HIP_DEVICE_COMPILE__) && defined(__gfx1250__) && \
    __has_builtin(__builtin_amdgcn_tensor_load_to_lds) &&      \
    __has_builtin(__builtin_amdgcn_s_wait_tensorcnt)
#define HAVE_TDM 1
#else
#define HAVE_TDM 0
#endif

#if HAVE_TDM
// DMA `nelems` fp32 from global to LDS byte offset `lds_off` via the TDM.
// D# per CDNA5 ISA ch.8: group0 {count=1, lds_addr, global_addr, type=2},
// group1 {data_size=4B, tensor_dim0=nelems, tensor_dim1=1, tile_dim0=nelems,
//         tile_dim1=1, dim0_stride=nelems}.
__device__ __forceinline__ void tdm_load_f32_row(const float* gsrc,
                                                 uint32_t lds_off,
                                                 uint32_t nelems) {
  uint64_t ga = (uint64_t)(uintptr_t)gsrc;
  v4u g0 = { 1u,                                   // count=1, user descriptor
             lds_off,                              // lds_addr
             (uint32_t)ga,                         // global_addr[31:0]
             (uint32_t)((ga >> 32) & 0x01FFFFFFu)  // global_addr[56:32]
                 | (2u << 30) };                   // type=2 ("image")
  v8i g1 = { (int)(2u << 16),          // data_size = 2 (4 bytes)
             (int)(nelems << 16),      // tensor_dim0[15:0]
             (int)(1u << 16),          // tensor_dim1 = 1
             (int)(nelems << 16),      // tile_dim0
             1,                        // tile_dim1 = 1
             (int)nelems,              // tensor_dim0_stride (low 32)
             0, 0 };
  v4i z4 = { 0, 0, 0, 0 };
#if defined(__clang_major__) && __clang_major__ >= 23
  v8i z8 = { 0, 0, 0, 0, 0, 0, 0, 0 };
  __builtin_amdgcn_tensor_load_to_lds(g0, g1, z4, z4, z8, 0);
#else
  __builtin_amdgcn_tensor_load_to_lds(g0, g1, z4, z4, 0);
#endif
  __builtin_amdgcn_s_wait_tensorcnt(0);
}
#endif

__device__ __forceinline__ uint32_t pcg_hash(uint32_t v) {
  uint32_t s = v * 747796405u + 2891336453u;
  uint32_t w = ((s >> ((s >> 28) + 4u)) ^ s) * 277803737u;
  return (w >> 22) ^ w;
}

__global__ __launch_bounds__(TPB) void harmonic_source_kernel(
    const float* __restrict__ f0, const float* __restrict__ W,
    const float* __restrict__ bias, float* __restrict__ out) {
  __shared__ float s_f0[FRAMES];        // whole f0 row, TDM-staged
  __shared__ float s_pref[FR_PER_BLK];  // exclusive phase prefix (revolutions)
  __shared__ float s_red[TPB / 32 + 1];

  const int tid   = threadIdx.x;
  const int chunk = blockIdx.x;  // 0..24
  const int row   = blockIdx.y;  // 0..31
  const int t0    = chunk * FR_PER_BLK;
  const int lane  = tid & 31;
  const int wid   = tid >> 5;
  const float* f0row = f0 + row * FRAMES;

  // ---- Stage f0 row into LDS (TDM on device pass; cooperative fallback) ----
#if HAVE_TDM
  if (wid == 0) {
    tdm_load_f32_row(f0row, (uint32_t)(uintptr_t)&s_f0[0], FRAMES);
  }
#else
  for (int i = tid; i < FRAMES; i += TPB) s_f0[i] = f0row[i];
#endif
  __syncthreads();

  // ---- Base phase: sum of f0 over all frames before this chunk ----
  float part = 0.0f;
  for (int i = tid; i < t0; i += TPB) part += s_f0[i];
  for (int d = 16; d > 0; d >>= 1) part += __shfl_down(part, d);
  if (lane == 0) s_red[wid] = part;
  __syncthreads();
  if (tid == 0) {
    float acc = 0.0f;
    for (int w = 0; w < TPB / 32; ++w) acc += s_red[w];
    s_red[TPB / 32] = acc;
  }
  __syncthreads();
  const float base = s_red[TPB / 32];

  // ---- Wave32 inclusive scan over this chunk's 32 frames -> excl. prefix ----
  if (wid == 0) {
    float v = s_f0[t0 + lane];
    float x = v;
    for (int d = 1; d < 32; d <<= 1) {
      float y = __shfl_up(x, d);
      if (lane >= d) x += y;
    }
    // revolutions: (sum f0[<t]) * upp / SR
    s_pref[lane] = (base + (x - v)) * ((float)FRAME_UP * SR_INV);
  }
  __syncthreads();

  // ---- Weights (uniform address -> scalar loads) ----
  const float w0 = W[0], w1 = W[1], w2 = W[2], w3 = W[3], w4 = W[4];
  const float w5 = W[5], w6 = W[6], w7 = W[7], w8 = W[8];
  const float bv = bias[0];

  float* __restrict__ out_har   = out;
  float* __restrict__ out_noise = out + STOT;
  float* __restrict__ out_uv    = out + 2 * STOT;
  const uint32_t rbase = (uint32_t)row * SAMPLES + (uint32_t)chunk * SAMP_PER_BLK;

  // phi is the base_phase in revolutions; sin(2*pi*k*phi) via v_sin + recurrence
  auto sample = [&](float phi) -> float {
    float s1 = __builtin_amdgcn_sinf(phi);     // sin(2*pi*phi)
    float c1 = __builtin_amdgcn_cosf(phi);
    float tc = 2.0f * c1;
    float acc = w0 * s1;
    float s2 = tc * s1;             acc = fmaf(w1, s2, acc);
    float s3 = fmaf(tc, s2, -s1);   acc = fmaf(w2, s3, acc);
    float s4 = fmaf(tc, s3, -s2);   acc = fmaf(w3, s4, acc);
    float s5 = fmaf(tc, s4, -s3);   acc = fmaf(w4, s5, acc);
    float s6 = fmaf(tc, s5, -s4);   acc = fmaf(w5, s6, acc);
    float s7 = fmaf(tc, s6, -s5);   acc = fmaf(w6, s7, acc);
    float s8 = fmaf(tc, s7, -s6);   acc = fmaf(w7, s8, acc);
    float s9 = fmaf(tc, s8, -s7);   acc = fmaf(w8, s9, acc);
    return acc;  // sum_k W[k] * sin(2*pi*(k+1)*phi)
  };

  auto tanh_fast = [](float x) -> float {
    x = fminf(fmaxf(x, -15.0f), 15.0f);
    float e = __builtin_amdgcn_exp2f(x * 2.8853900817779268f);  // exp(2x)
    return (e - 1.0f) * __builtin_amdgcn_rcpf(e + 1.0f);
  };

  auto gauss = [](uint32_t idx) -> float {
    uint32_t h1 = pcg_hash(idx * 2u + 0x9E3779B9u);
    uint32_t h2 = pcg_hash(idx * 2u + 1u);
    float u1 = (float)(h1 >> 8) * 0x1p-24f + 0x1p-25f;  // (0,1)
    float u2 = (float)(h2 >> 8) * 0x1p-24f;             // [0,1) revolutions
    float r  = __builtin_amdgcn_sqrtf(
        -1.3862943611198906f * __builtin_amdgcn_logf(u1));  // sqrt(-2 ln u1)
    return r * __builtin_amdgcn_cosf(u2) * NOISE_STD;
  };

#pragma unroll
  for (int it = 0; it < ITERS; ++it) {
    uint32_t q  = (uint32_t)(it * TPB + tid);  // float4 group id, 0..1919
    uint32_t tl = q / 60u;                     // frame within chunk (60 groups/frame)
    uint32_t sg = q - tl * 60u;                // group within frame

    float fv  = s_f0[t0 + tl];
    float uvv = (fv > 0.0f) ? 1.0f : 0.0f;
    float d   = fv * SR_INV;                                  // rev per sample
    float ph0 = s_pref[tl] + (float)(sg * 4u + 1u) * d;       // inclusive cumsum
    float amp = SINE_AMP * uvv;

    float4 h4;
    h4.x = tanh_fast(fmaf(sample(ph0),            amp, bv));
    h4.y = tanh_fast(fmaf(sample(ph0 + d),        amp, bv));
    h4.z = tanh_fast(fmaf(sample(ph0 + 2.0f * d), amp, bv));
    h4.w = tanh_fast(fmaf(sample(ph0 + 3.0f * d), amp, bv));

    uint32_t gi = rbase + q * 4u;
    float4 n4 = { gauss(gi), gauss(gi + 1u), gauss(gi + 2u), gauss(gi + 3u) };
    float4 u4 = { uvv, uvv, uvv, uvv };

    *(float4*)(out_har   + gi) = h4;   // global_store_b128, fully coalesced
    *(float4*)(out_noise + gi) = n4;
    *(float4*)(out_uv    + gi) = u4;
  }
}

extern "C" void kernel_launch(void* const* d_in, const int* in_sizes, int n_in,
                              void* d_out, int out_size, void* d_ws,
                              size_t ws_size, hipStream_t stream) {
  (void)in_sizes; (void)n_in; (void)out_size; (void)d_ws; (void)ws_size;
  const float* f0 = (const float*)d_in[0];
  const float* W  = (const float*)d_in[1];
  const float* b  = (const float*)d_in[2];
  // d_in[3] is upp (==240), fixed by the reference; hardcoded in the kernel.
  dim3 grid(FRAMES / FR_PER_BLK, BATCH);  // 25 x 32 blocks
  harmonic_source_kernel<<<grid, TPB, 0, stream>>>(f0, W, b, (float*)d_out);
}